// MLPSigmoid_67714454389334
// MI455X (gfx1250) — compile-verified
//
#include <hip/hip_runtime.h>
#include <cmath>

// ---------------------------------------------------------------------------
// Problem constants (from the reference):
//   X  : (B=4, N=1024, FMAP=64)  f32
//   W1 : (128, 32)  -> W1a = W1[0:64], W1b = W1[64:128]
//   A  = X @ W1a          (4,1024,32)
//   Cb = X @ W1b + b1     (4,1024,32)   (bias folded)
//   out[b,i,j] = sigmoid( sum_h tanh(A[b,j,h] + Cb[b,i,h]) * W2[h] + b2 )
// ---------------------------------------------------------------------------

#define FMAP 64
#define HID  32
#define NB   4
#define NN   1024
#define MTOT (NB * NN)   // 4096 rows

typedef __attribute__((ext_vector_type(2))) float v2f;
typedef __attribute__((ext_vector_type(8))) float v8f;

// ---- fast transcendentals -------------------------------------------------

__device__ __forceinline__ float fast_tanh(float x) {
#if __has_builtin(__builtin_amdgcn_tanhf)
    return __builtin_amdgcn_tanhf(x);
#else
    float r;
    asm volatile("v_tanh_f32 %0, %1" : "=v"(r) : "v"(x));
    return r;
#endif
}

__device__ __forceinline__ float fast_sigmoid(float x) {
    // sigmoid(x) = 0.5 * tanh(x/2) + 0.5  -> one TRANS op + one FMA
    return fmaf(0.5f, fast_tanh(0.5f * x), 0.5f);
}

// ---------------------------------------------------------------------------
// Phase 1: A = X@W1a, Cb = X@W1b + b1 using V_WMMA_F32_16X16X4_F32.
// One wave (32 threads) per 16-row tile of X. K=64 -> 16 steps of K=4.
// N=32 -> two 16-wide n-tiles, for each of the two weight halves.
//
// f32 A-operand layout (16x4):  lane L holds M = L%16, K = 2*(L/16)+v (v=VGPR)
// f32 B-operand layout (4x16):  lane L, VGPR v holds K = v + 2*(L/16), N = L%16
// f32 C/D layout (16x16):       lane L, VGPR r holds M = r + 8*(L/16), N = L%16
// ---------------------------------------------------------------------------
__global__ __launch_bounds__(32)
void mlp_phase1_wmma(const float* __restrict__ X,
                     const float* __restrict__ W1,
                     const float* __restrict__ b1,
                     float* __restrict__ Aout,
                     float* __restrict__ Cbout) {
    const int m0    = blockIdx.x * 16;       // row tile base
    const int lane  = threadIdx.x;           // 0..31
    const int mrow  = lane & 15;
    const int khalf = lane >> 4;             // 0 or 1
    const int ncol  = lane & 15;

    v8f accA0 = {}; v8f accA1 = {};          // A, n-tiles h=[0,16), [16,32)
    v8f accC0 = {}; v8f accC1 = {};          // C, n-tiles

    const float* xrow = X + (size_t)(m0 + mrow) * FMAP;

#pragma unroll
    for (int kk = 0; kk < 16; ++kk) {
        const int f0 = kk * 4 + 2 * khalf;   // this lane's K base within step

        v2f a;
        a.x = xrow[f0 + 0];
        a.y = xrow[f0 + 1];

        // B operands: W1a rows f0/f0+1, W1b rows 64+f0/64+f0+1
        v2f bA0, bA1, bC0, bC1;
        bA0.x = W1[(f0 + 0) * HID + ncol];
        bA0.y = W1[(f0 + 1) * HID + ncol];
        bA1.x = W1[(f0 + 0) * HID + 16 + ncol];
        bA1.y = W1[(f0 + 1) * HID + 16 + ncol];
        bC0.x = W1[(FMAP + f0 + 0) * HID + ncol];
        bC0.y = W1[(FMAP + f0 + 1) * HID + ncol];
        bC1.x = W1[(FMAP + f0 + 0) * HID + 16 + ncol];
        bC1.y = W1[(FMAP + f0 + 1) * HID + 16 + ncol];

        accA0 = __builtin_amdgcn_wmma_f32_16x16x4_f32(false, a, false, bA0,
                                                      (short)0, accA0, false, false);
        accA1 = __builtin_amdgcn_wmma_f32_16x16x4_f32(false, a, false, bA1,
                                                      (short)0, accA1, false, false);
        accC0 = __builtin_amdgcn_wmma_f32_16x16x4_f32(false, a, false, bC0,
                                                      (short)0, accC0, false, false);
        accC1 = __builtin_amdgcn_wmma_f32_16x16x4_f32(false, a, false, bC1,
                                                      (short)0, accC1, false, false);
    }

    const float b1lo = b1[ncol];
    const float b1hi = b1[16 + ncol];

#pragma unroll
    for (int r = 0; r < 8; ++r) {
        const int m   = r + 8 * khalf;
        const int row = m0 + m;
        Aout [row * HID + ncol]       = accA0[r];
        Aout [row * HID + 16 + ncol]  = accA1[r];
        Cbout[row * HID + ncol]       = accC0[r] + b1lo;
        Cbout[row * HID + 16 + ncol]  = accC1[r] + b1hi;
    }
}

// ---------------------------------------------------------------------------
// Phase 2: pairwise tanh + weighted reduce + sigmoid (compute-bound).
// Grid: (N/32 j-tiles, B, 8 i-groups), block = 256 (8 waves).
// Each wave: lane = one j; A[b,j,:] held in 32 VGPRs, reused over 16 i values.
// Cb row, W2, b2 are wave-uniform -> scalar loads.
// ---------------------------------------------------------------------------
__global__ __launch_bounds__(256)
void mlp_phase2_pairwise(const float* __restrict__ A,
                         const float* __restrict__ Cb,
                         const float* __restrict__ W2,
                         const float* __restrict__ b2,
                         float* __restrict__ out) {
    const int lane = threadIdx.x & 31;
    const int wave = threadIdx.x >> 5;
    const int b    = blockIdx.y;
    const int j    = blockIdx.x * 32 + lane;
    const int i0   = (blockIdx.z * 8 + wave) * 16;   // 16 i's per wave

    // Per-lane A row (32 f32 in VGPRs), vector loads.
    float regA[HID];
    {
        const float4* a4 = (const float4*)(A + (size_t)(b * NN + j) * HID);
#pragma unroll
        for (int q = 0; q < HID / 4; ++q) {
            float4 v = a4[q];
            regA[4 * q + 0] = v.x;
            regA[4 * q + 1] = v.y;
            regA[4 * q + 2] = v.z;
            regA[4 * q + 3] = v.w;
        }
    }

    // Wave-uniform W2 (compiler should place these via scalar loads).
    float w2[HID];
#pragma unroll
    for (int h = 0; h < HID; ++h) w2[h] = W2[h];
    const float bias2 = b2[0];

    float* orow = out + (size_t)(b * NN + i0) * NN + blockIdx.x * 32 + lane;

    for (int ii = 0; ii < 16; ++ii) {
        const float* crow = Cb + (size_t)(b * NN + i0 + ii) * HID;  // uniform addr
        float acc = 0.0f;
#pragma unroll
        for (int h = 0; h < HID; ++h) {
            const float t = fast_tanh(regA[h] + crow[h]);
            acc = fmaf(t, w2[h], acc);
        }
        *orow = fast_sigmoid(acc + bias2);
        orow += NN;
    }
}

// ---------------------------------------------------------------------------

extern "C" void kernel_launch(void* const* d_in, const int* in_sizes, int n_in,
                              void* d_out, int out_size, void* d_ws, size_t ws_size,
                              hipStream_t stream) {
    const float* X  = (const float*)d_in[0];
    const float* W1 = (const float*)d_in[1];
    const float* b1 = (const float*)d_in[2];
    const float* W2 = (const float*)d_in[3];
    const float* b2 = (const float*)d_in[4];
    float* out = (float*)d_out;

    // Workspace layout: A (4096x32 f32) | Cb (4096x32 f32)  = 1 MB total.
    float* A  = (float*)d_ws;
    float* Cb = A + (size_t)MTOT * HID;

    // Phase 1: 4096/16 = 256 row tiles, one wave each.
    mlp_phase1_wmma<<<dim3(MTOT / 16), dim3(32), 0, stream>>>(X, W1, b1, A, Cb);

    // Phase 2: 32 j-tiles x 4 batches x 8 i-groups, 8 waves per block.
    mlp_phase2_pairwise<<<dim3(NN / 32, NB, 8), dim3(256), 0, stream>>>(A, Cb, W2, b2, out);
}